// GraphSAGE_55731495633222
// MI455X (gfx1250) — compile-verified
//
#include <hip/hip_runtime.h>

// GraphSAGE 'gcn' aggregator, 2 layers, f32 throughout.
// d_in: feats[N*256], src[E], dst[E], W1[256*256], b1[256], W2[256*128], b2[128]
// d_out: h1 [N,256] followed by h2 [N,128]
// d_ws:  agg [N,256] f32 + deg [N] f32  (~103 MB)

typedef __attribute__((ext_vector_type(2))) float v2f;
typedef __attribute__((ext_vector_type(8))) float v8f;

#define DIN 256

// ---------------------------------------------------------------- zero
__global__ void zero_f32(float* __restrict__ p, size_t n) {
  size_t t = (size_t)blockIdx.x * blockDim.x + threadIdx.x;
  size_t stride = (size_t)gridDim.x * blockDim.x;
  for (; t < n; t += stride) p[t] = 0.0f;
}

// ---------------------------------------------------- edge scatter-add
// One thread per (edge, 4-float chunk): float4 gather from feats[src],
// 4 hardware f32 atomic adds into agg[dst] (L2-resident), deg count.
__global__ void scatter_edges(const float* __restrict__ feats,
                              const int* __restrict__ src,
                              const int* __restrict__ dst,
                              float* __restrict__ agg,
                              float* __restrict__ deg,
                              long long nwork) {
  long long t = (long long)blockIdx.x * blockDim.x + threadIdx.x;
  if (t >= nwork) return;
  const int e  = (int)(t >> 6);          // 64 chunks of 4 floats = 256
  const int kq = ((int)(t & 63)) << 2;
  const int s = src[e];
  const int d = dst[e];
  const float4 v = *(const float4*)(feats + (size_t)s * DIN + kq);
  float* p = agg + (size_t)d * DIN + kq;
  unsafeAtomicAdd(p + 0, v.x);
  unsafeAtomicAdd(p + 1, v.y);
  unsafeAtomicAdd(p + 2, v.z);
  unsafeAtomicAdd(p + 3, v.w);
  if ((t & 63) == 0) unsafeAtomicAdd(deg + d, 1.0f);
}

// -------------------------------------- h = (agg + self) / (deg + 1)
__global__ void normalize_inplace(float* __restrict__ agg,
                                  const float* __restrict__ selff,
                                  const float* __restrict__ deg,
                                  long long nwork) {
  long long t = (long long)blockIdx.x * blockDim.x + threadIdx.x;
  if (t >= nwork) return;
  const int i  = (int)(t >> 6);
  const int kq = ((int)(t & 63)) << 2;
  const float r = 1.0f / (deg[i] + 1.0f);
  float4 a = *(float4*)(agg + (size_t)i * DIN + kq);
  const float4 f = *(const float4*)(selff + (size_t)i * DIN + kq);
  a.x = (a.x + f.x) * r;
  a.y = (a.y + f.y) * r;
  a.z = (a.z + f.z) * r;
  a.w = (a.w + f.w) * r;
  *(float4*)(agg + (size_t)i * DIN + kq) = a;
}

// ------------------------------------------------- WMMA f32 GEMM
// out[N,Dout] = A[N,K] @ W[K,Dout] + bias, optional ReLU.
// 8 waves/block; each wave computes a 16x64 slab (4 accumulators) so the
// A fragment is reused across 4 N-tiles per v_wmma_f32_16x16x4_f32 step.
// A frag layout (ISA 16x4 f32): lane<16 -> K+0/K+1, lane>=16 -> K+2/K+3.
// B/C layout: row striped across lanes; VGPR v holds M=v (lo half) / v+8.
__global__ void wmma_gemm(const float* __restrict__ A,
                          const float* __restrict__ W,
                          const float* __restrict__ bias,
                          float* __restrict__ out,
                          int rowTiles, int K, int Dout, int doRelu) {
  const int lane = threadIdx.x & 31;
  const int wave = threadIdx.x >> 5;
  const int rt = blockIdx.x * 8 + wave;
  if (rt >= rowTiles) return;              // wave-uniform: EXEC stays all-1s

  const int row0 = rt * 16;
  const int col0 = blockIdx.y * 64;
  const int m  = lane & 15;                // row within tile (A) / col (B,C)
  const int hi = lane >> 4;                // K-half select for A/B fragments

  v8f acc[4] = {};
  const float* arow = A + (size_t)(row0 + m) * K + 2 * hi;

  for (int k = 0; k < K; k += 4) {
    const v2f a = *(const v2f*)(arow + k);            // A[m][k+2hi .. +1]
    const float* wp = W + (size_t)(k + 2 * hi) * Dout + col0 + m;
    v2f b[4];
#pragma unroll
    for (int tt = 0; tt < 4; ++tt) {
      b[tt][0] = wp[tt * 16];                         // K = k + 2hi
      b[tt][1] = wp[tt * 16 + Dout];                  // K = k + 2hi + 1
    }
#pragma unroll
    for (int tt = 0; tt < 4; ++tt) {
      acc[tt] = __builtin_amdgcn_wmma_f32_16x16x4_f32(
          /*neg_a=*/false, a, /*neg_b=*/false, b[tt],
          /*c_mod=*/(short)0, acc[tt], /*reuse_a=*/false, /*reuse_b=*/false);
    }
  }

#pragma unroll
  for (int tt = 0; tt < 4; ++tt) {
    const int col = col0 + tt * 16 + m;
    const float bv = bias[col];
#pragma unroll
    for (int v = 0; v < 8; ++v) {
      float x = acc[tt][v] + bv;
      if (doRelu) x = fmaxf(x, 0.0f);
      out[(size_t)(row0 + v + 8 * hi) * Dout + col] = x;
    }
  }
}

// ---------------------------------------------------------------- launch
extern "C" void kernel_launch(void* const* d_in, const int* in_sizes, int n_in,
                              void* d_out, int out_size, void* d_ws, size_t ws_size,
                              hipStream_t stream) {
  const float* feats = (const float*)d_in[0];
  const int*   src   = (const int*)d_in[1];
  const int*   dst   = (const int*)d_in[2];
  const float* W1    = (const float*)d_in[3];
  const float* b1    = (const float*)d_in[4];
  const float* W2    = (const float*)d_in[5];
  const float* b2    = (const float*)d_in[6];

  const int N = in_sizes[0] / DIN;     // 100000
  const int E = in_sizes[1];           // 1600000
  const int DHID = in_sizes[4];        // 256
  const int DOUT = in_sizes[6];        // 128

  float* h1 = (float*)d_out;                       // [N, DHID]
  float* h2 = h1 + (size_t)N * DHID;               // [N, DOUT]
  float* agg = (float*)d_ws;                       // [N, DIN]
  float* deg = agg + (size_t)N * DIN;              // [N]

  const size_t    zeroN   = (size_t)N * DIN + (size_t)N;
  const long long edgeW   = (long long)E * 64;     // (edge, 4-float chunk)
  const long long normW   = (long long)N * 64;
  const int edgeBlocks = (int)((edgeW + 255) / 256);
  const int normBlocks = (int)((normW + 255) / 256);
  const int rowTiles   = N / 16;                   // N % 16 == 0 here
  const int gemmBlocksX = (rowTiles + 7) / 8;

  // ---- layer 1: aggregate feats -> normalize -> GEMM(W1)+ReLU -> h1
  zero_f32<<<2048, 256, 0, stream>>>(agg, zeroN);
  scatter_edges<<<edgeBlocks, 256, 0, stream>>>(feats, src, dst, agg, deg, edgeW);
  normalize_inplace<<<normBlocks, 256, 0, stream>>>(agg, feats, deg, normW);
  {
    dim3 grid(gemmBlocksX, DHID / 64);
    wmma_gemm<<<grid, 256, 0, stream>>>(agg, W1, b1, h1, rowTiles, DIN, DHID, 1);
  }

  // ---- layer 2: aggregate h1 -> normalize -> GEMM(W2) -> h2
  zero_f32<<<2048, 256, 0, stream>>>(agg, zeroN);
  scatter_edges<<<edgeBlocks, 256, 0, stream>>>(h1, src, dst, agg, deg, edgeW);
  normalize_inplace<<<normBlocks, 256, 0, stream>>>(agg, h1, deg, normW);
  {
    dim3 grid(gemmBlocksX, DOUT / 64);
    wmma_gemm<<<grid, 256, 0, stream>>>(agg, W2, b2, h2, rowTiles, DHID, DOUT, 0);
  }
}